// MemoryAccess_9002251453169
// MI455X (gfx1250) — compile-verified
//
#include <hip/hip_runtime.h>
#include <math.h>

#define BB 8
#define TT 32
#define II 512
#define MM 1024
#define WDIM 64
#define RR 4
#define EPSF 1e-6f

typedef __attribute__((ext_vector_type(2))) float v2f;
typedef __attribute__((ext_vector_type(8))) float v8f;

__device__ __forceinline__ float dev_sigmoid(float x) { return 1.0f / (1.0f + __expf(-x)); }
__device__ __forceinline__ float dev_softplus(float x) { return x > 20.0f ? x : log1pf(__expf(x)); }

// ---------------------------------------------------------------------------
// Dense head GEMM: Y[256 x DOUT] = X[256 x 512] @ W[512 x DOUT] + b, + act.
// One wave per 16x16 output tile, fp32 WMMA 16x16x4.
// Branch-free inner loop: clamped column + 0/1 lane mask (no EXEC churn, so
// the WMMA all-lanes requirement holds throughout and the loop unrolls x8).
// ACT: 0 = none, 1 = sigmoid, 2 = softplus
// ---------------------------------------------------------------------------
template <int DOUT, int ACT>
__global__ void head_gemm(const float* __restrict__ X, const float* __restrict__ W,
                          const float* __restrict__ bias, float* __restrict__ Y) {
    const int lane = threadIdx.x & 31;
    const int half = lane >> 4;      // A: K pair select; C/D: rows M / M+8
    const int lm   = lane & 15;
    const int rowBase = blockIdx.x * 16;
    const int colBase = blockIdx.y * 16;
    const int col  = colBase + lm;
    const bool valid = (DOUT >= 16) ? true : (col < DOUT);
    const int colc = (DOUT >= 16) ? col : (valid ? col : (DOUT - 1));
    const float bmask = valid ? 1.0f : 0.0f;

    const float* arow = X + (size_t)(rowBase + lm) * II + 2 * half;
    const float* wcol = W + colc + (size_t)(2 * half) * DOUT;

    v8f acc = {};
#pragma unroll 8
    for (int k0 = 0; k0 < II; k0 += 4) {
        v2f a; a.x = arow[k0]; a.y = arow[k0 + 1];
        v2f b;
        b.x = wcol[(size_t)k0 * DOUT] * bmask;
        b.y = wcol[(size_t)(k0 + 1) * DOUT] * bmask;
        acc = __builtin_amdgcn_wmma_f32_16x16x4_f32(false, a, false, b,
                                                    (short)0, acc, false, false);
    }
    if (valid) {
        const float bv = bias[col];
#pragma unroll
        for (int r = 0; r < 8; ++r) {
            const int row = rowBase + r + 8 * half;
            float v = acc[r] + bv;
            if (ACT == 1) v = dev_sigmoid(v);
            else if (ACT == 2) v = dev_softplus(v);
            Y[(size_t)row * DOUT + col] = v;
        }
    }
}

// ---------------------------------------------------------------------------
// Content similarity GEMM: Dot[b, row, m] = sum_w Keys[b,row,w] * Mem[b,m,w]
// B-matrix is memory^T; fp32 WMMA 16x16x4, fully unrolled K=64 (16 WMMAs).
// ---------------------------------------------------------------------------
__global__ void key_dot_gemm(const float* __restrict__ Keys, const float* __restrict__ Mem,
                             float* __restrict__ Dot, int Tk) {
    const int lane = threadIdx.x & 31;
    const int half = lane >> 4;
    const int lm   = lane & 15;
    const int b = blockIdx.z;
    const int rowBase = blockIdx.x * 16;
    const int colBase = blockIdx.y * 16;

    const float* arow = Keys + (size_t)(b * Tk + rowBase + lm) * WDIM + 2 * half;
    const float* brow = Mem  + (size_t)(b * MM + colBase + lm) * WDIM + 2 * half;

    v8f acc = {};
#pragma unroll
    for (int k0 = 0; k0 < WDIM; k0 += 4) {
        v2f a;  a.x = arow[k0];  a.y = arow[k0 + 1];
        v2f bb; bb.x = brow[k0]; bb.y = brow[k0 + 1];
        acc = __builtin_amdgcn_wmma_f32_16x16x4_f32(false, a, false, bb,
                                                    (short)0, acc, false, false);
    }
#pragma unroll
    for (int r = 0; r < 8; ++r) {
        const int row = rowBase + r + 8 * half;
        Dot[(size_t)(b * Tk + row) * MM + colBase + lm] = acc[r];
    }
}

// ---------------------------------------------------------------------------
// Memory row norms over memory0: Mn[b*M+m] = ||Mem[b,m,:]||
// ---------------------------------------------------------------------------
__global__ void mem_norms(const float* __restrict__ Mem, float* __restrict__ Mn) {
    const int i = blockIdx.x * blockDim.x + threadIdx.x;
    if (i >= BB * MM) return;
    const float* p = Mem + (size_t)i * WDIM;
    float s = 0.0f;
#pragma unroll 4
    for (int w = 0; w < WDIM; ++w) { const float v = p[w]; s = fmaf(v, v, s); }
    Mn[i] = sqrtf(s);
}

// ---------------------------------------------------------------------------
// Cosine-similarity scaling + softmax over M=1024 (in-place on Dot).
// One 256-thread block per (b,row); 4 elements per thread.
// ---------------------------------------------------------------------------
__global__ void content_softmax(float* __restrict__ Dot, const float* __restrict__ Keys,
                                const float* __restrict__ Str, const float* __restrict__ Mn,
                                int Tk) {
    const int b   = blockIdx.x / Tk;
    const int row = blockIdx.x % Tk;
    const int tid = threadIdx.x;
    __shared__ float red[256];
    __shared__ float sh_kn;

    float pk = 0.0f;
    if (tid < WDIM) { const float kv = Keys[(size_t)(b * Tk + row) * WDIM + tid]; pk = kv * kv; }
    red[tid] = pk; __syncthreads();
    for (int s = 128; s > 0; s >>= 1) { if (tid < s) red[tid] += red[tid + s]; __syncthreads(); }
    if (tid == 0) sh_kn = sqrtf(red[0]);
    __syncthreads();
    const float kn = sh_kn;
    const float st = Str[b * Tk + row];

    float lv[4];
    float mx = -1e30f;
#pragma unroll
    for (int i = 0; i < 4; ++i) {
        const int m = tid + i * 256;
        const float d = Dot[(size_t)(b * Tk + row) * MM + m];
        const float sim = d / (kn * Mn[b * MM + m] + EPSF);
        lv[i] = sim * st;
        mx = fmaxf(mx, lv[i]);
    }
    __syncthreads();
    red[tid] = mx; __syncthreads();
    for (int s = 128; s > 0; s >>= 1) { if (tid < s) red[tid] = fmaxf(red[tid], red[tid + s]); __syncthreads(); }
    mx = red[0]; __syncthreads();

    float sum = 0.0f;
#pragma unroll
    for (int i = 0; i < 4; ++i) { lv[i] = __expf(lv[i] - mx); sum += lv[i]; }
    red[tid] = sum; __syncthreads();
    for (int s = 128; s > 0; s >>= 1) { if (tid < s) red[tid] += red[tid + s]; __syncthreads(); }
    const float inv = 1.0f / red[0];
#pragma unroll
    for (int i = 0; i < 4; ++i)
        Dot[(size_t)(b * Tk + row) * MM + tid + i * 256] = lv[i] * inv;
}

// ---------------------------------------------------------------------------
// read_mode softmax over last dim (3), in-place. One thread per (b,t,r).
// ---------------------------------------------------------------------------
__global__ void readmode_softmax(float* __restrict__ Y) {
    const int i = blockIdx.x * blockDim.x + threadIdx.x;
    if (i >= BB * TT * RR) return;
    const float a = Y[i * 3 + 0], b = Y[i * 3 + 1], c = Y[i * 3 + 2];
    const float m = fmaxf(a, fmaxf(b, c));
    const float ea = __expf(a - m), eb = __expf(b - m), ec = __expf(c - m);
    const float inv = 1.0f / (ea + eb + ec);
    Y[i * 3 + 0] = ea * inv; Y[i * 3 + 1] = eb * inv; Y[i * 3 + 2] = ec * inv;
}

// ---------------------------------------------------------------------------
// Per-timestep: allocation (bitonic sort + prefix product in LDS), write
// weights, usage update, precedence update. One 1024-thread block per batch.
// ---------------------------------------------------------------------------
__global__ void step_alloc(const float* __restrict__ wcw, const float* __restrict__ rcw,
                           const float* __restrict__ wg, const float* __restrict__ ag,
                           const float* __restrict__ fg,
                           float* __restrict__ usage, float* __restrict__ ww,
                           const float* __restrict__ prec_in, float* __restrict__ prec_out,
                           int t) {
    const int b = blockIdx.x;
    const int tid = threadIdx.x;  // 1024 == MM
    __shared__ float sv[MM];
    __shared__ int   si[MM];
    __shared__ float sp[MM];

    const float u  = usage[b * MM + tid];
    const float ue = EPSF + (1.0f - EPSF) * u;
    sv[tid] = 1.0f - ue;   // nonusage
    si[tid] = tid;
    __syncthreads();

    // bitonic sort, descending by nonusage (top_k semantics)
    for (int k = 2; k <= MM; k <<= 1) {
        for (int j = k >> 1; j > 0; j >>= 1) {
            const int p = tid ^ j;
            if (p > tid) {
                const bool desc = ((tid & k) == 0);
                const float v0 = sv[tid], v1 = sv[p];
                if ((v0 < v1) == desc) {
                    sv[tid] = v1; sv[p] = v0;
                    const int t0 = si[tid]; si[tid] = si[p]; si[p] = t0;
                }
            }
            __syncthreads();
        }
    }

    // inclusive prefix product of sorted usage
    sp[tid] = 1.0f - sv[tid];
    __syncthreads();
    for (int off = 1; off < MM; off <<= 1) {
        const float x = (tid >= off) ? sp[tid - off] : 1.0f;
        __syncthreads();
        sp[tid] *= x;
        __syncthreads();
    }
    const float excl = (tid == 0) ? 1.0f : sp[tid - 1];
    const float alloc_sorted = sv[tid] * excl;
    __syncthreads();
    sp[si[tid]] = alloc_sorted;      // scatter back to memory-slot order
    __syncthreads();
    const float alloc_m = sp[tid];

    const float agv = ag[b * TT + t];
    const float wgv = wg[b * TT + t];
    const float wgs = agv * wgv;
    const float wcw_m = wcw[(size_t)(b * TT + t) * MM + tid];
    const float w = wgs * (agv * alloc_m + (1.0f - agv) * wcw_m);
    ww[b * MM + tid] = w;

    // usage update (NW==1: write_w == w)
    const float after = u + (1.0f - u) * w;
    float phi = 1.0f;
#pragma unroll
    for (int r = 0; r < RR; ++r) {
        const float f  = fg[(b * TT + t) * RR + r];
        const float rw = rcw[(size_t)((b * TT + t) * RR + r) * MM + tid];
        phi *= (1.0f - f * rw);
    }
    usage[b * MM + tid] = after * phi;

    // precedence: new_prec = (1 - sum(ww)) * prec + ww
    __syncthreads();
    sv[tid] = w;
    __syncthreads();
    for (int s = 512; s > 0; s >>= 1) { if (tid < s) sv[tid] += sv[tid + s]; __syncthreads(); }
    const float wsum = sv[0];
    prec_out[b * MM + tid] = (1.0f - wsum) * prec_in[b * MM + tid] + w;
}

// ---------------------------------------------------------------------------
// Memory erase + write: mem[b,m,w] = mem*(1 - ww[m]*ev[w]) + ww[m]*wv[w]
// ---------------------------------------------------------------------------
__global__ void step_mem(const float* __restrict__ ww, const float* __restrict__ ev,
                         const float* __restrict__ wv, float* __restrict__ mem, int t) {
    const int i = blockIdx.x * blockDim.x + threadIdx.x;
    if (i >= BB * MM * WDIM) return;
    const int w  = i % WDIM;
    const int bm = i / WDIM;
    const int b  = bm / MM;
    const float W = ww[bm];
    const float e = ev[(size_t)(b * TT + t) * WDIM + w];
    const float v = wv[(size_t)(b * TT + t) * WDIM + w];
    mem[i] = mem[i] * (1.0f - W * e) + W * v;
}

// ---------------------------------------------------------------------------
// Link update (in-place, elementwise, float4 over j):
// L[i,j] = ((1 - ww[i] - ww[j]) * L[i,j] + ww[i]*prec[j]) * (i != j)
// ---------------------------------------------------------------------------
__global__ void step_link(float* __restrict__ link, const float* __restrict__ ww,
                          const float* __restrict__ prec) {
    const size_t e = (size_t)blockIdx.x * blockDim.x + threadIdx.x;  // quad index
    const int qb = MM * MM / 4;
    const int b   = (int)(e / qb);
    const int rem = (int)(e % qb);
    const int i   = rem / (MM / 4);
    const int j   = (rem % (MM / 4)) * 4;

    const float wi = ww[b * MM + i];
    const float4 wj = *(const float4*)(ww + b * MM + j);
    const float4 pj = *(const float4*)(prec + b * MM + j);
    float4* lp = (float4*)(link + (size_t)b * MM * MM + (size_t)i * MM + j);
    float4 l = *lp;
    l.x = (i == j + 0) ? 0.0f : ((1.0f - wi - wj.x) * l.x + wi * pj.x);
    l.y = (i == j + 1) ? 0.0f : ((1.0f - wi - wj.y) * l.y + wi * pj.y);
    l.z = (i == j + 2) ? 0.0f : ((1.0f - wi - wj.z) * l.z + wi * pj.z);
    l.w = (i == j + 3) ? 0.0f : ((1.0f - wi - wj.w) * l.w + wi * pj.w);
    *lp = l;
}

// ---------------------------------------------------------------------------
// fwd[r,k] = sum_m rw[r,m]*L[k,m]; bwd[r,k] = sum_m rw[r,m]*L[m,k];
// new_rw[r,k] = cmode*rcw + fmode*fwd + bmode*bwd.
// Block: 256 threads = 4 read-heads x 64 k. read-weights staged in LDS.
// ---------------------------------------------------------------------------
__global__ void step_fwdbwd(const float* __restrict__ link, const float* __restrict__ rw_in,
                            const float* __restrict__ rcw, const float* __restrict__ rm,
                            float* __restrict__ rw_out, int t) {
    const int b  = blockIdx.x / (MM / 64);
    const int kb = (blockIdx.x % (MM / 64)) * 64;
    const int tid = threadIdx.x;
    const int r = tid >> 6;
    const int k = kb + (tid & 63);

    __shared__ float srw[RR * MM];
    for (int i = tid; i < RR * MM; i += 256) srw[i] = rw_in[(size_t)b * RR * MM + i];
    __syncthreads();

    const float* Lb   = link + (size_t)b * MM * MM;
    const float* Lrow = Lb + (size_t)k * MM;
    const float* srwr = srw + r * MM;
    float fwd = 0.0f, bwd = 0.0f;
    for (int m = 0; m < MM; m += 4) {
        if ((m & 63) == 0 && m + 64 < MM) __builtin_prefetch(Lrow + m + 64, 0, 0);
        const float4 l4 = *(const float4*)(Lrow + m);
        fwd = fmaf(srwr[m + 0], l4.x, fwd);
        fwd = fmaf(srwr[m + 1], l4.y, fwd);
        fwd = fmaf(srwr[m + 2], l4.z, fwd);
        fwd = fmaf(srwr[m + 3], l4.w, fwd);
        bwd = fmaf(srwr[m + 0], Lb[(size_t)(m + 0) * MM + k], bwd);
        bwd = fmaf(srwr[m + 1], Lb[(size_t)(m + 1) * MM + k], bwd);
        bwd = fmaf(srwr[m + 2], Lb[(size_t)(m + 2) * MM + k], bwd);
        bwd = fmaf(srwr[m + 3], Lb[(size_t)(m + 3) * MM + k], bwd);
    }
    const int base = (b * TT + t) * RR + r;
    const float bmode = rm[base * 3 + 0];
    const float fmode = rm[base * 3 + 1];
    const float cmode = rm[base * 3 + 2];
    const float c = rcw[(size_t)base * MM + k];
    rw_out[(size_t)(b * RR + r) * MM + k] = cmode * c + fmode * fwd + bmode * bwd;
}

// ---------------------------------------------------------------------------
// read_words[b,t,r,w] = sum_m rw[b,r,m] * mem[b,m,w]
// ---------------------------------------------------------------------------
__global__ void step_read(const float* __restrict__ rw, const float* __restrict__ mem,
                          float* __restrict__ out, int t) {
    const int b = blockIdx.x / RR;
    const int r = blockIdx.x % RR;
    const int w = threadIdx.x;  // 64
    const float* rwr = rw + (size_t)(b * RR + r) * MM;
    const float* mb  = mem + (size_t)b * MM * WDIM;
    float acc = 0.0f;
    for (int m = 0; m < MM; ++m) acc = fmaf(rwr[m], mb[(size_t)m * WDIM + w], acc);
    out[(size_t)((b * TT + t) * RR + r) * WDIM + w] = acc;
}

// ---------------------------------------------------------------------------
extern "C" void kernel_launch(void* const* d_in, const int* in_sizes, int n_in,
                              void* d_out, int out_size, void* d_ws, size_t ws_size,
                              hipStream_t stream) {
    (void)in_sizes; (void)n_in; (void)out_size; (void)ws_size;
    const float* x        = (const float*)d_in[0];
    const float* memory0  = (const float*)d_in[1];
    const float* read_w0  = (const float*)d_in[2];
    const float* link0    = (const float*)d_in[3];
    const float* prec_init= (const float*)d_in[4];
    const float* usage0   = (const float*)d_in[5];

    float* ws = (float*)d_ws;
    size_t off = 0;
    auto take = [&](size_t n) { float* p = ws + off; off += n; return p; };
    float* a_wv   = take((size_t)BB * TT * WDIM);
    float* a_ev   = take((size_t)BB * TT * WDIM);
    float* a_wg   = take((size_t)BB * TT);
    float* a_ag   = take((size_t)BB * TT);
    float* a_fg   = take((size_t)BB * TT * RR);
    float* a_rm   = take((size_t)BB * TT * RR * 3);
    float* a_wk   = take((size_t)BB * TT * WDIM);
    float* a_wstr = take((size_t)BB * TT);
    float* a_rk   = take((size_t)BB * TT * RR * WDIM);
    float* a_rstr = take((size_t)BB * TT * RR);
    float* wcw    = take((size_t)BB * TT * MM);
    float* rcw    = take((size_t)BB * TT * RR * MM);
    float* mn     = take((size_t)BB * MM);
    float* usage  = take((size_t)BB * MM);
    float* wwbuf  = take((size_t)BB * MM);
    float* prec0  = take((size_t)BB * MM);
    float* prec1  = take((size_t)BB * MM);
    float* rw0    = take((size_t)BB * RR * MM);
    float* rw1    = take((size_t)BB * RR * MM);
    float* mem    = take((size_t)BB * MM * WDIM);
    float* link   = take((size_t)BB * MM * MM);

    // mutable state init (inputs must not be mutated)
    hipMemcpyAsync(link, link0, (size_t)BB * MM * MM * sizeof(float), hipMemcpyDeviceToDevice, stream);
    hipMemcpyAsync(mem, memory0, (size_t)BB * MM * WDIM * sizeof(float), hipMemcpyDeviceToDevice, stream);
    hipMemcpyAsync(rw0, read_w0, (size_t)BB * RR * MM * sizeof(float), hipMemcpyDeviceToDevice, stream);
    hipMemcpyAsync(prec0, prec_init, (size_t)BB * MM * sizeof(float), hipMemcpyDeviceToDevice, stream);
    hipMemcpyAsync(usage, usage0, (size_t)BB * MM * sizeof(float), hipMemcpyDeviceToDevice, stream);

    // dense projections (WMMA fp32, templated on DOUT/ACT)
    const int ROWT = (BB * TT) / 16;
    head_gemm<WDIM,     0><<<dim3(ROWT, WDIM / 16),      32, 0, stream>>>(x, (const float*)d_in[ 6], (const float*)d_in[ 7], a_wv);
    head_gemm<WDIM,     1><<<dim3(ROWT, WDIM / 16),      32, 0, stream>>>(x, (const float*)d_in[ 8], (const float*)d_in[ 9], a_ev);
    head_gemm<1,        1><<<dim3(ROWT, 1),              32, 0, stream>>>(x, (const float*)d_in[10], (const float*)d_in[11], a_wg);
    head_gemm<1,        1><<<dim3(ROWT, 1),              32, 0, stream>>>(x, (const float*)d_in[12], (const float*)d_in[13], a_ag);
    head_gemm<RR,       1><<<dim3(ROWT, 1),              32, 0, stream>>>(x, (const float*)d_in[14], (const float*)d_in[15], a_fg);
    head_gemm<RR * 3,   0><<<dim3(ROWT, 1),              32, 0, stream>>>(x, (const float*)d_in[16], (const float*)d_in[17], a_rm);
    head_gemm<WDIM,     0><<<dim3(ROWT, WDIM / 16),      32, 0, stream>>>(x, (const float*)d_in[18], (const float*)d_in[19], a_wk);
    head_gemm<1,        2><<<dim3(ROWT, 1),              32, 0, stream>>>(x, (const float*)d_in[20], (const float*)d_in[21], a_wstr);
    head_gemm<RR * WDIM,0><<<dim3(ROWT, (RR*WDIM) / 16), 32, 0, stream>>>(x, (const float*)d_in[22], (const float*)d_in[23], a_rk);
    head_gemm<RR,       2><<<dim3(ROWT, 1),              32, 0, stream>>>(x, (const float*)d_in[24], (const float*)d_in[25], a_rstr);

    readmode_softmax<<<(BB * TT * RR + 255) / 256, 256, 0, stream>>>(a_rm);

    // content weights from memory0 (precomputed outside the scan, as in reference)
    mem_norms<<<(BB * MM + 255) / 256, 256, 0, stream>>>(memory0, mn);
    key_dot_gemm<<<dim3(TT / 16, MM / 16, BB), 32, 0, stream>>>(a_wk, memory0, wcw, TT);
    key_dot_gemm<<<dim3((TT * RR) / 16, MM / 16, BB), 32, 0, stream>>>(a_rk, memory0, rcw, TT * RR);
    content_softmax<<<BB * TT, 256, 0, stream>>>(wcw, a_wk, a_wstr, mn, TT);
    content_softmax<<<BB * TT * RR, 256, 0, stream>>>(rcw, a_rk, a_rstr, mn, TT * RR);

    // sequential scan over T
    for (int t = 0; t < TT; ++t) {
        float* pin  = (t & 1) ? prec1 : prec0;
        float* pout = (t & 1) ? prec0 : prec1;
        float* rin  = (t & 1) ? rw1 : rw0;
        float* rout = (t & 1) ? rw0 : rw1;
        step_alloc<<<BB, MM, 0, stream>>>(wcw, rcw, a_wg, a_ag, a_fg,
                                          usage, wwbuf, pin, pout, t);
        step_mem<<<(BB * MM * WDIM) / 256, 256, 0, stream>>>(wwbuf, a_ev, a_wv, mem, t);
        step_link<<<(BB * MM * MM / 4) / 256, 256, 0, stream>>>(link, wwbuf, pin);
        step_fwdbwd<<<BB * (MM / 64), 256, 0, stream>>>(link, rin, rcw, a_rm, rout, t);
        step_read<<<BB * RR, WDIM, 0, stream>>>(rout, mem, (float*)d_out, t);
    }
}